// spikeGelu6_1_20684562497586
// MI455X (gfx1250) — compile-verified
//
#include <hip/hip_runtime.h>

// K = 16 fixed recurrence steps (length of h/d/T in the reference).
#define K 16

typedef __attribute__((ext_vector_type(4))) float v4f;

// Streaming spike-recurrence kernel.
// Math note: reference computes z = ((v - T) / (|v| + 1) > 0). Since the
// denominator is strictly positive, the predicate is exactly (v > T): the
// divide is eliminated. z in {0,1} makes z*h and z*d exact, so FMA
// contraction is bit-identical to the reference's mul+sub / mul+add.
__global__ __launch_bounds__(256)
void snn16_stream_kernel(const float* __restrict__ x,
                         const float* __restrict__ hp,
                         const float* __restrict__ dp,
                         const float* __restrict__ Tp,
                         float* __restrict__ out,
                         long long n4)
{
    // Thread-invariant constants -> scalar loads / SGPRs.
    float h[K], d[K], T[K];
#pragma unroll
    for (int t = 0; t < K; ++t) {
        h[t] = hp[t];
        d[t] = dp[t];
        T[t] = Tp[t];
    }

    const v4f* __restrict__ x4 = (const v4f*)x;
    v4f* __restrict__ o4 = (v4f*)out;

    long long i = (long long)blockIdx.x * blockDim.x + threadIdx.x;
    const long long gsz = (long long)gridDim.x * blockDim.x;

    for (; i < n4; i += gsz) {
        // Deepen MLP: prefetch next grid-stride iteration (global_prefetch_b8).
        if (i + gsz < n4) {
            __builtin_prefetch(&x4[i + gsz], 0, 0);
        }

        // 128-bit non-temporal load: stream is single-use, > L2 capacity.
        v4f xv = __builtin_nontemporal_load(&x4[i]);

        float v0 = xv.x, v1 = xv.y, v2 = xv.z, v3 = xv.w;
        float z0 = 0.f, z1 = 0.f, z2 = 0.f, z3 = 0.f;
        float o0 = 0.f, o1 = 0.f, o2 = 0.f, o3 = 0.f;

#pragma unroll
        for (int t = 0; t < K; ++t) {
            // v -= z * h[t]   (z from previous step; t==0 folds away, z==0)
            v0 = __builtin_fmaf(-z0, h[t], v0);
            v1 = __builtin_fmaf(-z1, h[t], v1);
            v2 = __builtin_fmaf(-z2, h[t], v2);
            v3 = __builtin_fmaf(-z3, h[t], v3);
            // z = spike((v - T[t]) / (|v|+1))  ==  (v > T[t])
            z0 = (v0 > T[t]) ? 1.0f : 0.0f;
            z1 = (v1 > T[t]) ? 1.0f : 0.0f;
            z2 = (v2 > T[t]) ? 1.0f : 0.0f;
            z3 = (v3 > T[t]) ? 1.0f : 0.0f;
            // out += z * d[t]
            o0 = __builtin_fmaf(z0, d[t], o0);
            o1 = __builtin_fmaf(z1, d[t], o1);
            o2 = __builtin_fmaf(z2, d[t], o2);
            o3 = __builtin_fmaf(z3, d[t], o3);
        }

        v4f ov;
        ov.x = o0; ov.y = o1; ov.z = o2; ov.w = o3;
        __builtin_nontemporal_store(ov, &o4[i]);
    }
}

// Scalar tail (n % 4 != 0 — not hit for 4*4096*4096, kept for generality).
__global__ __launch_bounds__(64)
void snn16_tail_kernel(const float* __restrict__ x,
                       const float* __restrict__ hp,
                       const float* __restrict__ dp,
                       const float* __restrict__ Tp,
                       float* __restrict__ out,
                       long long start, long long n)
{
    long long i = start + (long long)blockIdx.x * blockDim.x + threadIdx.x;
    if (i >= n) return;
    float v = x[i], z = 0.f, o = 0.f;
#pragma unroll
    for (int t = 0; t < K; ++t) {
        v = __builtin_fmaf(-z, hp[t], v);
        z = (v > Tp[t]) ? 1.0f : 0.0f;
        o = __builtin_fmaf(z, dp[t], o);
    }
    out[i] = o;
}

extern "C" void kernel_launch(void* const* d_in, const int* in_sizes, int n_in,
                              void* d_out, int out_size, void* d_ws, size_t ws_size,
                              hipStream_t stream) {
    const float* x = (const float*)d_in[0];
    const float* h = (const float*)d_in[1];
    const float* d = (const float*)d_in[2];
    const float* T = (const float*)d_in[3];
    float* out = (float*)d_out;

    const long long n   = (long long)in_sizes[0];
    const long long n4  = n / 4;
    const long long rem = n % 4;

    const int block = 256;              // 8 wave32s per block
    const int itersPerThread = 4;       // amortize constant loads, keep MLP high
    long long threadsNeeded = (n4 + itersPerThread - 1) / itersPerThread;
    long long gridLL = (threadsNeeded + block - 1) / block;
    if (gridLL < 1) gridLL = 1;
    if (gridLL > 1048576) gridLL = 1048576;
    const int grid = (int)gridLL;

    if (n4 > 0) {
        snn16_stream_kernel<<<grid, block, 0, stream>>>(x, h, d, T, out, n4);
    }
    if (rem > 0) {
        snn16_tail_kernel<<<1, 64, 0, stream>>>(x, h, d, T, out, n4 * 4, n);
    }
}